// NMS_58497454571603
// MI455X (gfx1250) — compile-verified
//
#include <hip/hip_runtime.h>
#include <stdint.h>
#include <stddef.h>

namespace {
constexpr int   kB      = 8;
constexpr int   kN      = 25200;
constexpr int   kD      = 200;      // 5 + 15 + 180
constexpr int   kNC     = 15;
constexpr int   kNA     = 180;
constexpr int   kS      = 2048;     // PRE_NMS
constexpr int   kCap    = 4096;     // selection capacity before sort
constexpr int   kMaxDet = 300;
constexpr int   kNbin   = 65536;
constexpr float kConf   = 0.3f;
constexpr float kIouT   = 0.6f;
constexpr float kMaxWh  = 4096.0f;
}

// ---------------------------------------------------------------------------
// gfx1250 async global->LDS copy (ASYNCcnt path, CDNA5 ISA)
// ---------------------------------------------------------------------------
__device__ __forceinline__ void async_g2l_b128(uint32_t lds_byte_addr,
                                               uint32_t voffset,
                                               const void* sbase) {
  asm volatile("global_load_async_to_lds_b128 %0, %1, %2"
               :: "v"(lds_byte_addr), "v"(voffset), "s"(sbase)
               : "memory");
}
__device__ __forceinline__ void wait_async0() {
  asm volatile("s_wait_asynccnt 0" ::: "memory");
}
// low 32 bits of a flat LDS pointer == byte offset from LDS_BASE (ISA 10.2)
__device__ __forceinline__ uint32_t lds_byte_off(const void* p) {
  return (uint32_t)(uintptr_t)p;
}

// ---------------------------------------------------------------------------
// 1) init workspace
// ---------------------------------------------------------------------------
__global__ void nms_init(uint32_t* hist, float* candScore, uint32_t* candIdx,
                         int* cnt, uint32_t* kept) {
  int i = blockIdx.x * blockDim.x + threadIdx.x;
  int stride = gridDim.x * blockDim.x;
  for (int k = i; k < kB * kNbin; k += stride) hist[k] = 0u;
  for (int k = i; k < kB * kCap; k += stride) { candScore[k] = -1.0f; candIdx[k] = 0u; }
  for (int k = i; k < kB * 64; k += stride) kept[k] = 0u;
  for (int k = i; k < kB; k += stride) cnt[k] = 0;
}

// ---------------------------------------------------------------------------
// 2) score pass + float-bit histogram (radix-select prep)
// ---------------------------------------------------------------------------
__global__ void nms_score_hist(const float* __restrict__ pred,
                               uint32_t* __restrict__ hist) {
  int idx = blockIdx.x * blockDim.x + threadIdx.x;
  if (idx >= kB * kN) return;
  int b = idx / kN;
  const float* row = pred + (size_t)idx * kD;
  __builtin_prefetch(row + 256 * kD, 0, 0);   // global_prefetch_b8 (stream hint)
  float obj = row[4];
  if (obj <= kConf) return;
  uint32_t* h = hist + (size_t)b * kNbin;
  #pragma unroll
  for (int c = 0; c < kNC; ++c) {
    float s = obj * row[5 + c];
    if (s > kConf) {
      uint32_t bin = __float_as_uint(s) >> 16;   // positive floats: monotone
      atomicAdd(&h[bin], 1u);
    }
  }
}

// ---------------------------------------------------------------------------
// 3) per-batch threshold bin (suffix count >= kS)
// ---------------------------------------------------------------------------
__global__ void nms_threshold(const uint32_t* __restrict__ hist,
                              int* __restrict__ thrBin) {
  int b = threadIdx.x;
  if (b >= kB) return;
  const uint32_t* h = hist + (size_t)b * kNbin;
  unsigned cum = 0; int t = 0;
  for (int bin = kNbin - 1; bin >= 0; --bin) {
    cum += h[bin];
    if (cum >= (unsigned)kS) { t = bin; break; }
  }
  thrBin[b] = t;
}

// ---------------------------------------------------------------------------
// 4) compact candidates with bin >= threshold (cap kCap)
// ---------------------------------------------------------------------------
__global__ void nms_compact(const float* __restrict__ pred,
                            const int* __restrict__ thrBin,
                            float* __restrict__ candScore,
                            uint32_t* __restrict__ candIdx,
                            int* __restrict__ cnt) {
  int idx = blockIdx.x * blockDim.x + threadIdx.x;
  if (idx >= kB * kN) return;
  int b = idx / kN;
  int n = idx - b * kN;
  const float* row = pred + (size_t)idx * kD;
  __builtin_prefetch(row + 256 * kD, 0, 0);
  float obj = row[4];
  if (obj <= kConf) return;
  int T = thrBin[b];
  #pragma unroll
  for (int c = 0; c < kNC; ++c) {
    float s = obj * row[5 + c];
    if (s > kConf) {
      int bin = (int)(__float_as_uint(s) >> 16);
      if (bin >= T) {
        int pos = atomicAdd(&cnt[b], 1);
        if (pos < kCap) {
          candScore[(size_t)b * kCap + pos] = s;
          candIdx[(size_t)b * kCap + pos]   = (uint32_t)(n * 16 + c);
        }
      }
    }
  }
}

// ---------------------------------------------------------------------------
// 5) per-batch bitonic sort (desc) + gather boxes, theta argmax, geometry
// ---------------------------------------------------------------------------
__global__ __launch_bounds__(1024)
void nms_sort_finalize(const float* __restrict__ pred,
                       const float* __restrict__ candScore,
                       const uint32_t* __restrict__ candIdx,
                       float* __restrict__ det,     // [B][kS][7]
                       float* __restrict__ geom) {  // [B][5][kS] planar
  __shared__ float    skey[kCap];
  __shared__ uint32_t sval[kCap];
  int b = blockIdx.x;
  int tid = threadIdx.x;
  for (int i = tid; i < kCap; i += 1024) {
    skey[i] = candScore[(size_t)b * kCap + i];
    sval[i] = candIdx[(size_t)b * kCap + i];
  }
  __syncthreads();
  // bitonic sort, descending by key
  for (int kk = 2; kk <= kCap; kk <<= 1) {
    for (int j = kk >> 1; j > 0; j >>= 1) {
      for (int i = tid; i < kCap; i += 1024) {
        int ixj = i ^ j;
        if (ixj > i) {
          bool up = (i & kk) == 0;
          if ((skey[i] < skey[ixj]) == up) {
            float tf = skey[i]; skey[i] = skey[ixj]; skey[ixj] = tf;
            uint32_t tv = sval[i]; sval[i] = sval[ixj]; sval[ixj] = tv;
          }
        }
      }
      __syncthreads();
    }
  }
  // finalize top-kS
  for (int r = tid; r < kS; r += 1024) {
    float s = skey[r];
    uint32_t pv = sval[r];
    float cx = 0.f, cy = 0.f, wd = 0.f, ht = 0.f, th = 0.f, cls = 0.f;
    if (s > kConf) {
      int n = (int)(pv >> 4);
      int c = (int)(pv & 15u);
      const float* p = pred + ((size_t)b * kN + n) * kD;
      cx = p[0]; cy = p[1]; wd = p[2]; ht = p[3];
      int   am = 0;
      float bm = p[5 + kNC];
      for (int k = 1; k < kNA; ++k) {
        float t = p[5 + kNC + k];
        if (t > bm) { bm = t; am = k; }
      }
      th  = (float)(am - 90) * 0.017453292519943295f;
      cls = (float)c;
    } else {
      s = -1.0f;
    }
    float* dr = det + ((size_t)b * kS + r) * 7;
    dr[0] = cx; dr[1] = cy; dr[2] = wd; dr[3] = ht;
    dr[4] = th; dr[5] = s;  dr[6] = cls;
    float off = cls * kMaxWh;
    float ox = cx + off, oy = cy + off;
    float* g = geom + (size_t)b * 5 * kS;
    g[0 * kS + r] = ox - wd * 0.5f;
    g[1 * kS + r] = oy - ht * 0.5f;
    g[2 * kS + r] = ox + wd * 0.5f;
    g[3 * kS + r] = oy + ht * 0.5f;
    g[4 * kS + r] = wd * ht;
  }
}

// ---------------------------------------------------------------------------
// 6) suppress bit-matrix: async-stage 40KB geometry tile to LDS, 32 IoUs/thread
//    grid: (kB, kS/4), block 256 = 4 rows x 64 words
// ---------------------------------------------------------------------------
__global__ __launch_bounds__(256)
void nms_iou(const float* __restrict__ geom,
             uint32_t* __restrict__ sup) {           // [B][kS][64]
  __shared__ float sg[5 * kS];                        // 40 KB
  int b   = blockIdx.x;
  int tid = threadIdx.x;

  // async global -> LDS copy of this batch's geometry planes (2560 x 16B)
  {
    const char* gbase = (const char*)(geom + (size_t)b * 5 * kS);
    uint32_t lbase = lds_byte_off(&sg[0]);
    #pragma unroll
    for (int k = 0; k < 10; ++k) {
      uint32_t chunk = (uint32_t)(tid + k * 256);
      async_g2l_b128(lbase + chunk * 16u, chunk * 16u, gbase);
    }
    wait_async0();
  }
  __syncthreads();

  int word = tid & 63;
  int i    = blockIdx.y * 4 + (tid >> 6);
  float x1i = sg[0 * kS + i], y1i = sg[1 * kS + i];
  float x2i = sg[2 * kS + i], y2i = sg[3 * kS + i];
  float ai  = sg[4 * kS + i];

  uint32_t bits = 0u;
  int j0 = word * 32;
  #pragma unroll 8
  for (int t = 0; t < 32; ++t) {
    int j = j0 + t;
    float iw = fminf(x2i, sg[2 * kS + j]) - fmaxf(x1i, sg[0 * kS + j]);
    float ih = fminf(y2i, sg[3 * kS + j]) - fmaxf(y1i, sg[1 * kS + j]);
    iw = fmaxf(iw, 0.0f);
    ih = fmaxf(ih, 0.0f);
    float inter = iw * ih;
    float iou = inter / (ai + sg[4 * kS + j] - inter + 1e-7f);
    if (iou > kIouT) bits |= (1u << t);
  }
  sup[((size_t)b * kS + i) * 64 + word] = bits;
}

// ---------------------------------------------------------------------------
// 7) greedy scan: single wave32 per batch, keep mask in registers,
//    software-pipelined row loads, ballot for any(suppress & keep)
// ---------------------------------------------------------------------------
__global__ __launch_bounds__(32)
void nms_scan(const float* __restrict__ det,
              const uint32_t* __restrict__ sup,
              uint32_t* __restrict__ kept) {
  int t = threadIdx.x;          // 0..31, owns keep words 2t, 2t+1
  int b = blockIdx.x;

  // valid bits for candidates [64t, 64t+64)
  uint32_t vA = 0u, vB = 0u;
  for (int k = 0; k < 32; ++k) {
    int r = 64 * t + k;
    if (det[((size_t)b * kS + r) * 7 + 5] > kConf) vA |= (1u << k);
  }
  for (int k = 0; k < 32; ++k) {
    int r = 64 * t + 32 + k;
    if (det[((size_t)b * kS + r) * 7 + 5] > kConf) vB |= (1u << k);
  }

  uint32_t kA = 0u, kB_ = 0u;
  const uint32_t* rows = sup + (size_t)b * kS * 64;
  uint32_t c0 = rows[0 * 64 + 2 * t];
  uint32_t c1 = rows[0 * 64 + 2 * t + 1];

  for (int i = 0; i < kS; ++i) {
    uint32_t n0 = 0u, n1 = 0u;
    if (i + 1 < kS) {                    // prefetch next row (independent)
      n0 = rows[(size_t)(i + 1) * 64 + 2 * t];
      n1 = rows[(size_t)(i + 1) * 64 + 2 * t + 1];
    }
    uint32_t hit = (c0 & kA) | (c1 & kB_);
    bool any = (__ballot((int)(hit != 0u)) & 0xffffffffull) != 0ull;
    int ls = i >> 6;
    int wsel = (i >> 5) & 1;
    uint32_t vw = (uint32_t)__shfl((int)(wsel ? vB : vA), ls, 32);
    bool ki = (((vw >> (i & 31)) & 1u) != 0u) && !any;
    if (ki && t == ls) {
      if (wsel) kB_ |= (1u << (i & 31));
      else      kA  |= (1u << (i & 31));
    }
    c0 = n0; c1 = n1;
  }
  kept[b * 64 + 2 * t]     = kA;
  kept[b * 64 + 2 * t + 1] = kB_;
}

// ---------------------------------------------------------------------------
// 8) output: candidates already score-desc; first 300 kept == top-k of kept
// ---------------------------------------------------------------------------
__global__ void nms_output(const float* __restrict__ det,
                           const uint32_t* __restrict__ kept,
                           float* __restrict__ out) {
  int b = blockIdx.x;
  int tid = threadIdx.x;
  float* outDet  = out + (size_t)b * kMaxDet * 7;
  float* outMask = out + (size_t)kB * kMaxDet * 7 + (size_t)b * kMaxDet;
  for (int i = tid; i < kMaxDet * 7; i += blockDim.x) outDet[i] = 0.0f;
  for (int i = tid; i < kMaxDet; i += blockDim.x) outMask[i] = 0.0f;
  __syncthreads();
  if (tid == 0) {
    int c = 0;
    for (int i = 0; i < kS && c < kMaxDet; ++i) {
      if ((kept[b * 64 + (i >> 5)] >> (i & 31)) & 1u) {
        const float* dr = det + ((size_t)b * kS + i) * 7;
        #pragma unroll
        for (int q = 0; q < 7; ++q) outDet[c * 7 + q] = dr[q];
        outMask[c] = 1.0f;
        ++c;
      }
    }
  }
}

// ---------------------------------------------------------------------------
extern "C" void kernel_launch(void* const* d_in, const int* in_sizes, int n_in,
                              void* d_out, int out_size, void* d_ws, size_t ws_size,
                              hipStream_t stream) {
  (void)in_sizes; (void)n_in; (void)out_size; (void)ws_size;
  const float* pred = (const float*)d_in[0];
  float* out = (float*)d_out;

  char* w = (char*)d_ws;
  uint32_t* sup       = (uint32_t*)w;  w += (size_t)kB * kS * 64 * sizeof(uint32_t); // 4 MiB
  uint32_t* hist      = (uint32_t*)w;  w += (size_t)kB * kNbin * sizeof(uint32_t);   // 2 MiB
  float*    det       = (float*)w;     w += (size_t)kB * kS * 7 * sizeof(float);
  float*    geom      = (float*)w;     w += (size_t)kB * kS * 5 * sizeof(float);
  float*    candScore = (float*)w;     w += (size_t)kB * kCap * sizeof(float);
  uint32_t* candIdx   = (uint32_t*)w;  w += (size_t)kB * kCap * sizeof(uint32_t);
  uint32_t* kept      = (uint32_t*)w;  w += (size_t)kB * 64 * sizeof(uint32_t);
  int*      cnt       = (int*)w;       w += (size_t)kB * sizeof(int);
  int*      thrBin    = (int*)w;       w += (size_t)kB * sizeof(int);

  nms_init<<<1024, 256, 0, stream>>>(hist, candScore, candIdx, cnt, kept);

  int nbox = kB * kN;
  nms_score_hist<<<(nbox + 255) / 256, 256, 0, stream>>>(pred, hist);
  nms_threshold<<<1, kB, 0, stream>>>(hist, thrBin);
  nms_compact<<<(nbox + 255) / 256, 256, 0, stream>>>(pred, thrBin, candScore, candIdx, cnt);
  nms_sort_finalize<<<kB, 1024, 0, stream>>>(pred, candScore, candIdx, det, geom);
  nms_iou<<<dim3(kB, kS / 4), 256, 0, stream>>>(geom, sup);
  nms_scan<<<kB, 32, 0, stream>>>(det, sup, kept);
  nms_output<<<kB, 256, 0, stream>>>(det, kept, out);
}